// VPN_19731079758389
// MI455X (gfx1250) — compile-verified
//
#include <hip/hip_runtime.h>
#include <hip/hip_bf16.h>
#include <math.h>

// ---------------------------------------------------------------------------
// Problem constants (from reference)
// ---------------------------------------------------------------------------
#define HH 512
#define WW 512
#define NSTATE (HH * WW)            // 262144
#define NS3 (3 * NSTATE)            // 786432  (W1 cols)
#define NS4 (4 * NSTATE)            // 1048576 (Wpol cols)
#define HID1 32
#define HID2 64
#define HIDP 32
#define NACT 8
#define KITER 10

#define CHUNK_COLS 512              // K-chunk per wave for wide GEMVs
#define CH_W1  (NS3 / CHUNK_COLS)   // 1536
#define CH_POL (NS4 / CHUNK_COLS)   // 2048
#define PF_DIST 1024                // prefetch distance in floats (4 KB)

typedef float v2f __attribute__((ext_vector_type(2)));
typedef float v8f __attribute__((ext_vector_type(8)));

// ---------------------------------------------------------------------------
// WMMA helper: D = A(16x4 f32) * B(4x16 f32) + C   (V_WMMA_F32_16X16X4_F32)
// A layout: lane l (m=l&15, kh=l>>4): a.x=A[m][2*kh], a.y=A[m][2*kh+1]
// B layout: lane l: b.x=B[2*kh+0][l&15], b.y=B[2*kh+1][l&15]
// C/D layout: vgpr r, lanes 0-15: D[r][n], lanes 16-31: D[8+r][n]
// ---------------------------------------------------------------------------
__device__ __forceinline__ v8f wmma_f32_k4(v2f a, v2f b, v8f c) {
  return __builtin_amdgcn_wmma_f32_16x16x4_f32(
      /*neg_a=*/false, a, /*neg_b=*/false, b,
      /*c_mod=*/(short)0, c, /*reuse_a=*/false, /*reuse_b=*/false);
}

// ---------------------------------------------------------------------------
// Wide GEMV partials: rows=32 (2 row-tiles of 16), very long K.
// One wave (32 threads) per block. grid.x = 2 * numChunks.
// Input vector is concat(x0[0:n0], x1[0:cols-n0]); n0 is a multiple of
// CHUNK_COLS so the source is uniform per chunk (for W1 pass x1=x0, n0=cols).
// partial[row * numChunks + chunk] = sum over this K-chunk of W[row][k]*x[k]
// ---------------------------------------------------------------------------
__global__ void k_gemv_wide_partial(const float* __restrict__ W,
                                    const float* __restrict__ x0, int n0,
                                    const float* __restrict__ x1,
                                    int cols, int numChunks,
                                    float* __restrict__ partial) {
  const int chunk   = blockIdx.x % numChunks;
  const int rowTile = blockIdx.x / numChunks;
  const int lane = threadIdx.x;        // 0..31
  const int m    = lane & 15;          // row within tile
  const int kh   = lane >> 4;          // K half (0 or 1)
  const int koff = kh * 2;
  const int row  = rowTile * 16 + m;
  const int kbase = chunk * CHUNK_COLS;

  // Uniform per-chunk source select (n0 % CHUNK_COLS == 0 guaranteed).
  const float* __restrict__ xs = (kbase < n0) ? x0 : (x1 - n0);

  // Loop-invariant per-lane base pointers; inner loop uses small constant offsets.
  const float* __restrict__ wp = W + (size_t)row * (size_t)cols + kbase + koff;
  const float* __restrict__ xp = xs + kbase + koff;

  v8f c = {0.f, 0.f, 0.f, 0.f, 0.f, 0.f, 0.f, 0.f};
  for (int k0 = 0; k0 < CHUNK_COLS; k0 += 64) {
    // Speculative prefetch (RT-speculative TH: OOB translation silently
    // dropped per ISA §10.5) — one per 16 WMMAs.
    __builtin_prefetch(wp + k0 + PF_DIST, 0, 1);   // global_prefetch_b8
    #pragma unroll
    for (int k = 0; k < 64; k += 4) {
      const v2f a  = *(const v2f*)(wp + k0 + k);   // global_load_b64
      const v2f xv = *(const v2f*)(xp + k0 + k);   // broadcast pair
      v2f b;
      b.x = (m == 0) ? xv.x : 0.f;
      b.y = (m == 0) ? xv.y : 0.f;
      c = wmma_f32_k4(a, b, c);
    }
  }
  // Column N=0 of D lives in lanes 0 (M=0..7) and 16 (M=8..15).
  if (m == 0) {
    const int mbase = rowTile * 16 + kh * 8;
    #pragma unroll
    for (int r = 0; r < 8; ++r)
      partial[(size_t)(mbase + r) * numChunks + chunk] = c[r];
  }
}

// Deterministic fixed-order reduction of partials + bias + relu.
__global__ void k_reduce_bias_relu(const float* __restrict__ partial,
                                   const float* __restrict__ bias,
                                   float* __restrict__ out,
                                   int rows, int numChunks) {
  const int r = blockIdx.x * blockDim.x + threadIdx.x;
  if (r >= rows) return;
  const float* p = partial + (size_t)r * numChunks;
  float s = 0.f;
  for (int c = 0; c < numChunks; ++c) s += p[c];
  s += bias[r];
  out[r] = s > 0.f ? s : 0.f;
}

// h2 = relu(W2 @ h1 + b2)   (64 x 32, one 64-thread block)
__global__ void k_h2(const float* __restrict__ W2, const float* __restrict__ b2,
                     const float* __restrict__ h1, float* __restrict__ h2) {
  const int r = threadIdx.x;   // 0..63
  float s = b2[r];
  #pragma unroll
  for (int k = 0; k < HID1; ++k) s += W2[r * HID1 + k] * h1[k];
  h2[r] = s > 0.f ? s : 0.f;
}

// ---------------------------------------------------------------------------
// Three fused sigmoid GEMVs: out[row] = sigmoid(W[row,:64] . h2 + b[row])
// block = 256 (8 waves), each wave -> 16 rows via 16 WMMA_F32_16x16x4 steps.
// gridDim.x = NSTATE/128 = 2048, gridDim.y = 3 (selects W/b/out).
// ---------------------------------------------------------------------------
__global__ void k_sig_gemv(const float* __restrict__ Wa, const float* __restrict__ Ba, float* __restrict__ Oa,
                           const float* __restrict__ Wb, const float* __restrict__ Bb, float* __restrict__ Ob,
                           const float* __restrict__ Wc, const float* __restrict__ Bc, float* __restrict__ Oc,
                           const float* __restrict__ h2) {
  const float* W; const float* B; float* O;
  if (blockIdx.y == 0)      { W = Wa; B = Ba; O = Oa; }
  else if (blockIdx.y == 1) { W = Wb; B = Bb; O = Ob; }
  else                      { W = Wc; B = Bc; O = Oc; }

  const int wave = threadIdx.x >> 5;
  const int lane = threadIdx.x & 31;
  const int m    = lane & 15;
  const int kh   = lane >> 4;
  const int koff = kh * 2;
  const int rowbase = blockIdx.x * 128 + wave * 16;

  const float* __restrict__ wp = W + (size_t)(rowbase + m) * HID2 + koff;
  const float* __restrict__ hp = h2 + koff;

  v8f c = {0.f, 0.f, 0.f, 0.f, 0.f, 0.f, 0.f, 0.f};
  #pragma unroll
  for (int k = 0; k < HID2; k += 4) {
    const v2f a  = *(const v2f*)(wp + k);    // global_load_b64
    const v2f hv = *(const v2f*)(hp + k);
    v2f b;
    b.x = (m == 0) ? hv.x : 0.f;
    b.y = (m == 0) ? hv.y : 0.f;
    c = wmma_f32_k4(a, b, c);
  }
  if (m == 0) {
    const int mb = rowbase + kh * 8;
    #pragma unroll
    for (int r = 0; r < 8; ++r) {
      const float s = c[r] + B[mb + r];
      O[mb + r] = 1.f / (1.f + expf(-s));
    }
  }
}

// rd = rin - rout (in place into rin buffer); v init to 0
__global__ void k_prep(float* __restrict__ rin_to_rd,
                       const float* __restrict__ rout,
                       float* __restrict__ v0) {
  const int i = blockIdx.x * blockDim.x + threadIdx.x;
  if (i < NSTATE) {
    rin_to_rd[i] = rin_to_rd[i] - rout[i];
    v0[i] = 0.f;
  }
}

// One value-iteration step:
// vout[i,j] = max over 8 neighbor offsets d of (v[n]*p[n] + rd[n]) with the
// jnp.pad(...,1) semantics: out-of-bounds candidate is exactly 0.0.
__global__ void k_step(const float* __restrict__ v, const float* __restrict__ p,
                       const float* __restrict__ rd, float* __restrict__ vout) {
  const int j = blockIdx.x * blockDim.x + threadIdx.x;  // col
  const int i = blockIdx.y;                             // row
  if (j >= WW) return;
  float best = -3.402823466e38f;
  #pragma unroll
  for (int di = -1; di <= 1; ++di) {
    #pragma unroll
    for (int dj = -1; dj <= 1; ++dj) {
      if (di == 0 && dj == 0) continue;
      const int ni = i + di, nj = j + dj;
      float cand;
      if (ni >= 0 && ni < HH && nj >= 0 && nj < WW) {
        const int n = ni * WW + nj;
        cand = fmaf(v[n], p[n], rd[n]);
      } else {
        cand = 0.f;   // padded: 0*0 + 0 - 0
      }
      best = fmaxf(best, cand);
    }
  }
  vout[i * WW + j] = best;
}

// logits = Whead @ hpol + bhead ; softmax ; state_value = v[pos]
__global__ void k_head(const float* __restrict__ Whead, const float* __restrict__ bhead,
                       const float* __restrict__ hpol, const float* __restrict__ v,
                       const int* __restrict__ pos, float* __restrict__ out) {
  __shared__ float logits[NACT];
  const int t = threadIdx.x;
  if (t < NACT) {
    float s = bhead[t];
    #pragma unroll
    for (int k = 0; k < HIDP; ++k) s += Whead[t * HIDP + k] * hpol[k];
    logits[t] = s;
  }
  __syncthreads();
  if (t == 0) {
    float mx = logits[0];
    for (int i = 1; i < NACT; ++i) mx = fmaxf(mx, logits[i]);
    float e[NACT]; float sum = 0.f;
    for (int i = 0; i < NACT; ++i) { e[i] = expf(logits[i] - mx); sum += e[i]; }
    for (int i = 0; i < NACT; ++i) out[i] = e[i] / sum;
    out[NACT] = v[pos[0] * WW + pos[1]];
  }
}

// ---------------------------------------------------------------------------
// Host-side orchestration
// ---------------------------------------------------------------------------
extern "C" void kernel_launch(void* const* d_in, const int* in_sizes, int n_in,
                              void* d_out, int out_size, void* d_ws, size_t ws_size,
                              hipStream_t stream) {
  (void)in_sizes; (void)n_in; (void)out_size; (void)ws_size;

  const float* x     = (const float*)d_in[0];   // (3,512,512)
  const int*   pos   = (const int*)  d_in[1];   // (2,)
  const float* W1    = (const float*)d_in[2];   // (32, 786432)
  const float* b1    = (const float*)d_in[3];
  const float* W2    = (const float*)d_in[4];   // (64, 32)
  const float* b2    = (const float*)d_in[5];
  const float* Wro   = (const float*)d_in[6];   // (262144, 64)
  const float* bro   = (const float*)d_in[7];
  const float* Wri   = (const float*)d_in[8];
  const float* bri   = (const float*)d_in[9];
  const float* Wp    = (const float*)d_in[10];
  const float* bp    = (const float*)d_in[11];
  const float* Wpol  = (const float*)d_in[12];  // (32, 1048576)
  const float* bpol  = (const float*)d_in[13];
  const float* Whead = (const float*)d_in[14];  // (8, 32)
  const float* bhead = (const float*)d_in[15];
  float* out = (float*)d_out;                   // 8 probs + 1 value

  // workspace carve-up (floats)
  float* ws    = (float*)d_ws;
  float* p     = ws;                   // 262144
  float* rd    = ws + NSTATE;          // rin -> rd (262144)
  float* rout  = ws + 2 * NSTATE;      // 262144
  float* vA    = ws + 3 * NSTATE;      // 262144
  float* vB    = ws + 4 * NSTATE;      // 262144
  float* h1    = ws + 5 * NSTATE;      // 32
  float* h2    = h1 + HID1;            // 64
  float* hpol  = h2 + HID2;            // 32
  float* part  = hpol + HIDP;          // up to 32*2048

  // 1) h1 = relu(W1 @ state + b1)  (WMMA partials + deterministic reduce)
  k_gemv_wide_partial<<<dim3(2 * CH_W1), dim3(32), 0, stream>>>(
      W1, x, NS3, x, NS3, CH_W1, part);
  k_reduce_bias_relu<<<dim3(1), dim3(32), 0, stream>>>(part, b1, h1, HID1, CH_W1);

  // 2) h2 = relu(W2 @ h1 + b2)
  k_h2<<<dim3(1), dim3(HID2), 0, stream>>>(W2, b2, h1, h2);

  // 3) rout/rin/p = sigmoid(W @ h2 + b)  (fused, WMMA)
  k_sig_gemv<<<dim3(NSTATE / 128, 3), dim3(256), 0, stream>>>(
      Wro, bro, rout,
      Wri, bri, rd,     // rin goes into rd buffer, then converted in-place
      Wp,  bp,  p,
      h2);

  // 4) rd = rin - rout ; v = 0
  k_prep<<<dim3((NSTATE + 255) / 256), dim3(256), 0, stream>>>(rd, rout, vA);

  // 5) K value-iteration steps (ping-pong)
  float* cur = vA;
  float* nxt = vB;
  for (int it = 0; it < KITER; ++it) {
    k_step<<<dim3((WW + 255) / 256, HH), dim3(256), 0, stream>>>(cur, p, rd, nxt);
    float* t = cur; cur = nxt; nxt = t;
  }
  // KITER is even -> final v in vA (== cur)

  // 6) hpol = relu(Wpol @ concat(v, state) + bpol)
  k_gemv_wide_partial<<<dim3(2 * CH_POL), dim3(32), 0, stream>>>(
      Wpol, cur, NSTATE, x, NS4, CH_POL, part);
  k_reduce_bias_relu<<<dim3(1), dim3(32), 0, stream>>>(part, bpol, hpol, HIDP, CH_POL);

  // 7) head: softmax(Whead @ hpol + bhead), value = v[pos]
  k_head<<<dim3(1), dim3(32), 0, stream>>>(Whead, bhead, hpol, cur, pos, out);
}